// ModulationLoss_53300544143767
// MI455X (gfx1250) — compile-verified
//
#include <hip/hip_runtime.h>

// ---------------------------------------------------------------------------
// ModulationLoss for MI455X (gfx1250, wave32).
// HBM-bound streaming kernel (~103 MB once -> ~4.4us roofline @ 23.3 TB/s).
// Branchless 8-corner trilinear gather (clamped addrs + masked weights) so the
// loads issue as one clause; WMMA f32 16x16x4 (ones-matrix) provides the exact
// f32 cross-lane reduction on the CDNA5 matrix path.
// ---------------------------------------------------------------------------

typedef __attribute__((ext_vector_type(2))) float v2f;
typedef __attribute__((ext_vector_type(8))) float v8f;

// Full wave32 sum via two V_WMMA_F32_16X16X4_F32 with B = ones.
// WMMA1: D[m][n] = x[m] + x[m+16]; per-lane sum of the 8 D regs gives the
// low/high half-sums; WMMA2 on those leaves the full 32-lane sum in d2[0].
__device__ __forceinline__ float wave_sum_wmma(float x) {
  v2f a; a[0] = x;    a[1] = 0.0f;
  v2f b; b[0] = 1.0f; b[1] = 1.0f;
  v8f c = {};
  v8f d = __builtin_amdgcn_wmma_f32_16x16x4_f32(
      /*neg_a=*/false, a, /*neg_b=*/false, b,
      /*c_mod=*/(short)0, c, /*reuse_a=*/false, /*reuse_b=*/false);
  float t = (d[0] + d[1]) + (d[2] + d[3]);
  t += (d[4] + d[5]) + (d[6] + d[7]);
  v2f a2; a2[0] = t; a2[1] = 0.0f;
  v8f d2 = __builtin_amdgcn_wmma_f32_16x16x4_f32(
      false, a2, false, b, (short)0, c, false, false);
  return d2[0];
}

// Problem constants (from reference): N=2, K=27, D=H=W=48, GT=96.
#define DIMS 48
#define HW   (48 * 48)          // 2304
#define DHW  (48 * 48 * 48)     // 110592
#define NTOT 5971968            // 2*27*48^3

__device__ __forceinline__ int clamp47(int v) {
  return min(max(v, 0), 47);
}

// Pass 1: one block per (n, k, z) slab; 256 threads cover the 48x48 slice
// in 9 exact iterations (no divergence at the WMMA reduction point).
__global__ __launch_bounds__(256)
void bce_partial_kernel(const float* __restrict__ offset,
                        const float* __restrict__ modulation,
                        const float* __restrict__ gt,
                        float* __restrict__ partial) {
  const int s = blockIdx.x;            // 0 .. 2*27*48-1
  const int z = s % DIMS;
  const int k = (s / DIMS) % 27;
  const int n = s / (DIMS * 27);

  // taps = product((-1,0,1), repeat=3) ordered (kz, ky, kx), kz slowest
  const float kx = (float)((k % 3) - 1);
  const float ky = (float)(((k / 3) % 3) - 1);
  const float kz = (float)((k / 9) - 1);

  const float* __restrict__ modp =
      modulation + (((size_t)(n * 27 + k) * DIMS + z) * HW);
  const float* __restrict__ offp =
      offset + (((size_t)(n * 81 + k * 3) * DIMS + z) * HW);
  const float* __restrict__ gtn = gt + (size_t)n * (96 * 96 * 96);

  const float sc = 49.0f / 50.0f;      // (1+sg) * 49/50 = padded-vol coordinate
  float acc = 0.0f;

  for (int i = threadIdx.x; i < HW; i += 256) {
    const int y = i / DIMS;
    const int x = i - y * DIMS;

    const float m  = modp[i];
    const float ox = offp[i];
    const float oy = offp[i + DHW];
    const float oz = offp[i + 2 * DHW];

    const float px = (1.0f + (float)x + kx + ox) * sc;
    const float py = (1.0f + (float)y + ky + oy) * sc;
    const float pz = (1.0f + (float)z + kz + oz) * sc;

    const float xf = floorf(px), yf = floorf(py), zf = floorf(pz);
    const float fx = px - xf, fy = py - yf, fz = pz - zf;
    const int x0 = (int)xf, y0 = (int)yf, z0 = (int)zf;

    // Padded 50^3 volume corner (zi,yi,xi) maps to gt[2(zi-1)][2(yi-1)][2(xi-1)]
    // when 1 <= idx <= 48, else contributes 0 (pad ring + zeros padding).
    // Branchless: clamp the address per axis, zero the *weight* when invalid
    // (exactly the reference's `wgt * valid`). All 8 loads issue as a clause.
    const int zi0 = z0 - 1, yi0 = y0 - 1, xi0 = x0 - 1;  // dz/dy/dx = 0 corner
    const float wz0 = ((unsigned)zi0 < 48u) ? (1.0f - fz) : 0.0f;
    const float wz1 = ((unsigned)z0  < 48u) ? fz          : 0.0f;
    const float wy0 = ((unsigned)yi0 < 48u) ? (1.0f - fy) : 0.0f;
    const float wy1 = ((unsigned)y0  < 48u) ? fy          : 0.0f;
    const float wx0 = ((unsigned)xi0 < 48u) ? (1.0f - fx) : 0.0f;
    const float wx1 = ((unsigned)x0  < 48u) ? fx          : 0.0f;

    const int az0 = 2 * clamp47(zi0) * (96 * 96);
    const int az1 = 2 * clamp47(z0)  * (96 * 96);
    const int ay0 = 2 * clamp47(yi0) * 96;
    const int ay1 = 2 * clamp47(y0)  * 96;
    const int ax0 = 2 * clamp47(xi0);
    const int ax1 = 2 * clamp47(x0);

    const float v000 = gtn[az0 + ay0 + ax0];
    const float v001 = gtn[az0 + ay0 + ax1];
    const float v010 = gtn[az0 + ay1 + ax0];
    const float v011 = gtn[az0 + ay1 + ax1];
    const float v100 = gtn[az1 + ay0 + ax0];
    const float v101 = gtn[az1 + ay0 + ax1];
    const float v110 = gtn[az1 + ay1 + ax0];
    const float v111 = gtn[az1 + ay1 + ax1];

    const float sv =
        wz0 * (wy0 * (wx0 * v000 + wx1 * v001) +
               wy1 * (wx0 * v010 + wx1 * v011)) +
        wz1 * (wy0 * (wx0 * v100 + wx1 * v101) +
               wy1 * (wx0 * v110 + wx1 * v111));

    // BCEWithLogits, numerically stable
    acc += fmaxf(m, 0.0f) - m * sv + __logf(1.0f + __expf(-fabsf(m)));
  }

  // exact-f32 wave32 reduction on the WMMA path
  const float wsum = wave_sum_wmma(acc);

  __shared__ float lsum[8];
  if ((threadIdx.x & 31) == 0) lsum[threadIdx.x >> 5] = wsum;
  __syncthreads();
  if (threadIdx.x == 0) {
    float b = 0.0f;
#pragma unroll
    for (int w = 0; w < 8; ++w) b += lsum[w];
    partial[blockIdx.x] = b;
  }
}

// Pass 2: single block, deterministic reduction of per-block partials.
__global__ __launch_bounds__(256)
void bce_reduce_kernel(const float* __restrict__ partial, int nPart,
                       float* __restrict__ out, float invTot) {
  float acc = 0.0f;
  for (int i = threadIdx.x; i < nPart; i += 256) acc += partial[i];

  const float wsum = wave_sum_wmma(acc);

  __shared__ float lsum[8];
  if ((threadIdx.x & 31) == 0) lsum[threadIdx.x >> 5] = wsum;
  __syncthreads();
  if (threadIdx.x == 0) {
    float b = 0.0f;
#pragma unroll
    for (int w = 0; w < 8; ++w) b += lsum[w];
    out[0] = b * invTot;
  }
}

extern "C" void kernel_launch(void* const* d_in, const int* in_sizes, int n_in,
                              void* d_out, int out_size, void* d_ws, size_t ws_size,
                              hipStream_t stream) {
  const float* offset     = (const float*)d_in[0];  // (2, 81, 48,48,48)
  const float* modulation = (const float*)d_in[1];  // (2, 27, 48,48,48)
  const float* gt_mask    = (const float*)d_in[2];  // (2, 96, 96, 96)
  float* out     = (float*)d_out;                   // scalar f32
  float* partial = (float*)d_ws;                    // 2592 floats of scratch

  const int nBlocks = 2 * 27 * 48;                  // one (n,k,z) slab each
  bce_partial_kernel<<<nBlocks, 256, 0, stream>>>(offset, modulation, gt_mask,
                                                  partial);
  bce_reduce_kernel<<<1, 256, 0, stream>>>(partial, nBlocks, out,
                                           1.0f / (float)NTOT);
}